// _normal_2d_30726196035749
// MI455X (gfx1250) — compile-verified
//
#include <hip/hip_runtime.h>

// 2x2 softplus-weighted pooling over x[16,128,256,256] f32 -> out[16,128,128,128] f32.
// Memory-bound streaming kernel: 512 MiB in + 128 MiB out (~29us floor @ 23.3 TB/s).
// Uses CDNA5 async global->LDS copies (ASYNCcnt) to double-buffer the input stream.

typedef float v4f __attribute__((ext_vector_type(4)));
typedef int   v4i __attribute__((ext_vector_type(4)));

#define THREADS 256
#define ITERS 8

#if defined(__gfx1250__) && __has_builtin(__builtin_amdgcn_global_load_async_to_lds_b128)
#define USE_ASYNC 1
#else
#define USE_ASYNC 0
#endif

#if USE_ASYNC
// Wait until at most n async ops remain, then force a compiler-level memory
// barrier so the LDS readback loads cannot be reordered above the wait.
#if __has_builtin(__builtin_amdgcn_s_wait_asynccnt)
#define WAIT_ASYNC(n)                          \
    do {                                       \
        __builtin_amdgcn_s_wait_asynccnt(n);   \
        asm volatile("" ::: "memory");         \
    } while (0)
#else
#define WAIT_ASYNC(n)                                          \
    do {                                                       \
        asm volatile("s_wait_asynccnt " #n ::: "memory");      \
    } while (0)
#endif
#endif

__device__ __forceinline__ float sp_w(float x) {
    // torch Softplus(threshold=10): x>10 -> x, else log(1+exp(min(x,10))).
    // t <= 10 so exp(t) <= 22026: no overflow; underflow -> log(1)=0 (correct limit).
    float t  = fminf(x, 10.0f);
    float sp = __logf(1.0f + __expf(t));
    return (x > 10.0f) ? x : sp;
}

__device__ __forceinline__ float pool4(float a, float b, float c, float d) {
    float wa = sp_w(a), wb = sp_w(b), wc = sp_w(c), wd = sp_w(d);
    float num = fmaf(a, wa, fmaf(b, wb, fmaf(c, wc, d * wd)));
    float den = (wa + wb) + (wc + wd);
    return __fdividef(num, den);
}

__device__ __forceinline__ v4f quad_compute(v4f r00, v4f r01, v4f r10, v4f r11) {
    // r00/r01: 8 consecutive floats of input row 2i; r10/r11: row 2i+1.
    v4f o;
    o.x = pool4(r00.x, r00.y, r10.x, r10.y);
    o.y = pool4(r00.z, r00.w, r10.z, r10.w);
    o.z = pool4(r01.x, r01.y, r11.x, r11.y);
    o.w = pool4(r01.z, r01.w, r11.z, r11.w);
    return o;
}

__global__ __launch_bounds__(THREADS) void pool2x2_kernel(const float* __restrict__ in,
                                                          float* __restrict__ out,
                                                          unsigned nquads) {
    const unsigned tid = threadIdx.x;
    v4f* __restrict__ outv = (v4f*)out;

#if USE_ASYNC
    // Double-buffered staging: 2 bufs x 256 threads x 4 x b128 = 32 KB LDS.
    __shared__ v4f sbuf[2][THREADS * 4];
    typedef v4i __attribute__((address_space(3)))* lds_p;
    typedef v4i __attribute__((address_space(1)))* glb_p;
    lds_p sb  = (lds_p)&sbuf[0][0];
    glb_p gin = (glb_p)in;

    unsigned chunk = blockIdx.x;
    const unsigned stride = gridDim.x;
    int buf = 0;

    // Per-quad addressing (v4 units): row_id = q>>5 (one output row = 32 quads),
    // row0 base = row_id*128 + (q&31)*2, row1 = +64.
#define ISSUE(ch, bi)                                                              \
    do {                                                                           \
        unsigned _q0 = (ch) * THREADS; /* nquads is a multiple of THREADS here */  \
        if (_q0 < nquads) {                                                        \
            unsigned _q  = _q0 + tid;                                              \
            unsigned _g0 = (_q >> 5) * 128u + (_q & 31u) * 2u;                     \
            lds_p _d = sb + (bi) * (THREADS * 4) + tid * 4;                        \
            __builtin_amdgcn_global_load_async_to_lds_b128(gin + _g0,      _d + 0, 0, 0); \
            __builtin_amdgcn_global_load_async_to_lds_b128(gin + _g0 + 1,  _d + 1, 0, 0); \
            __builtin_amdgcn_global_load_async_to_lds_b128(gin + _g0 + 64, _d + 2, 0, 0); \
            __builtin_amdgcn_global_load_async_to_lds_b128(gin + _g0 + 65, _d + 3, 0, 0); \
        }                                                                          \
    } while (0)

    ISSUE(chunk, 0);
    for (int it = 0; it < ITERS; ++it) {
        unsigned nxt = chunk + stride;
        bool prefetch = (it + 1 < ITERS) && (nxt * THREADS < nquads);
        if (prefetch) {
            ISSUE(nxt, buf ^ 1);
            WAIT_ASYNC(4);  // newest 4 = prefetch; older (current chunk) complete
        } else {
            WAIT_ASYNC(0);
        }
        unsigned q = chunk * THREADS + tid;
        if (q < nquads) {
            const v4f* s = (const v4f*)&sbuf[buf][tid * 4];  // wave-private: no barrier
            v4f r00 = s[0], r01 = s[1], r10 = s[2], r11 = s[3];
            __builtin_nontemporal_store(quad_compute(r00, r01, r10, r11), outv + q);
        }
        chunk = nxt;
        buf ^= 1;
    }
#undef ISSUE
#else
    const v4f* __restrict__ gin = (const v4f*)in;
    for (unsigned q = blockIdx.x * THREADS + tid; q < nquads;
         q += gridDim.x * THREADS) {
        unsigned g0 = (q >> 5) * 128u + (q & 31u) * 2u;
        v4f r00 = __builtin_nontemporal_load(gin + g0);
        v4f r01 = __builtin_nontemporal_load(gin + g0 + 1);
        v4f r10 = __builtin_nontemporal_load(gin + g0 + 64);
        v4f r11 = __builtin_nontemporal_load(gin + g0 + 65);
        __builtin_nontemporal_store(quad_compute(r00, r01, r10, r11), outv + q);
    }
#endif
}

extern "C" void kernel_launch(void* const* d_in, const int* in_sizes, int n_in,
                              void* d_out, int out_size, void* d_ws, size_t ws_size,
                              hipStream_t stream) {
    (void)in_sizes; (void)n_in; (void)d_ws; (void)ws_size;
    const float* x = (const float*)d_in[0];
    float* out = (float*)d_out;

    unsigned nquads = (unsigned)(out_size / 4);          // 4 outputs per thread-chunk
    unsigned chunks = (nquads + THREADS - 1) / THREADS;  // 256-thread chunks
    unsigned grid   = (chunks + ITERS - 1) / ITERS;      // each block pipelines ITERS chunks
    if (grid == 0) grid = 1;

    pool2x2_kernel<<<grid, THREADS, 0, stream>>>(x, out, nquads);
}